// ColBERT_15006615733213
// MI455X (gfx1250) — compile-verified
//
#include <hip/hip_runtime.h>

typedef __attribute__((ext_vector_type(2))) float v2f;
typedef __attribute__((ext_vector_type(8))) float v8f;

#define B_   64
#define Q_   32
#define LD_  4096
#define D_   128
#define QPAD 132          // query LDS row stride (floats): conflict-free b64 frag reads
#define KC   32           // K-chunk (dims) staged per async step
#define DPAD 36           // doc LDS row stride (floats): conflict-free + 16B-aligned rows
#define TOK  128          // doc tokens per block
#define DBUF_BYTES (TOK * DPAD * 4)

// Order-preserving float<->uint encoding so we can use integer atomicMax.
__device__ __forceinline__ unsigned encf(float f) {
    unsigned u = __float_as_uint(f);
    return (u >> 31) ? ~u : (u | 0x80000000u);
}
__device__ __forceinline__ float decf(unsigned s) {
    unsigned u = (s >> 31) ? (s & 0x7FFFFFFFu) : ~s;
    return __uint_as_float(u);
}

__global__ void colbert_init_ws(unsigned* __restrict__ ws) {
    int i = blockIdx.x * blockDim.x + threadIdx.x;
    if (i < B_ * Q_) ws[i] = 0u;   // 0 encodes below every real float
}

__global__ __launch_bounds__(256)
void colbert_maxsim(const float* __restrict__ q,
                    const float* __restrict__ d,
                    unsigned* __restrict__ ws) {
    __shared__ float sQ[Q_ * QPAD];          // 16.9 KB queries (full K)
    __shared__ float sD[2][TOK * DPAD];      // 36.9 KB double-buffered doc K-chunks
    __shared__ float sMax[8][Q_];            // per-wave per-query tile maxima

    const int b    = blockIdx.y;
    const int tid  = threadIdx.x;
    const int wave = tid >> 5;
    const int lane = tid & 31;

    const int half = lane >> 4;   // ISA: lanes 16-31 carry K = base+2, base+3
    const int mrow = lane & 15;   // A: M row.  B: N column (doc token in tile)
    const int tok0 = wave * 16;   // this wave's 16 doc tokens within the block tile
    const int tsub = lane >> 3;   // async copy: token sub-index (0..3)
    const int dsub = lane & 7;    // async copy: 16B sub-chunk within 128B row slice

    // Global base of this block's 128-token doc tile (bytes).
    const char* dtile =
        (const char*)(d + ((size_t)b * LD_ + (size_t)blockIdx.x * TOK) * D_);
    // LDS byte offset of sD (generic LDS pointers: low 32 bits = LDS offset).
    const unsigned ldsD = (unsigned)(unsigned long long)(void*)&sD[0][0];

    // Issue one K-chunk (this wave's 16 tokens x 32 dims = 2KB) as 4 coalesced
    // async B128 copies: 32 lanes x 16B = 512B/instr = 4 fully-used cache lines.
    auto issue_chunk = [&](int kc, int buf) {
#pragma unroll
        for (int j = 0; j < 4; ++j) {
            const int t = tok0 + j * 4 + tsub;
            unsigned long long ga = (unsigned long long)dtile
                                  + (unsigned)(t * (D_ * 4) + kc * (KC * 4) + dsub * 16);
            unsigned la = ldsD + (unsigned)(buf * DBUF_BYTES + t * (DPAD * 4) + dsub * 16);
            asm volatile("global_load_async_to_lds_b128 %0, %1, off"
                         :: "v"(la), "v"(ga) : "memory");
        }
    };

    // Prime the pipeline: chunk 0 -> buffer 0 (overlaps with query staging).
    issue_chunk(0, 0);

    // Stage this batch's queries into LDS (coalesced).
    const float* qb = q + (size_t)b * Q_ * D_;
    for (int i = tid; i < Q_ * D_; i += 256) {
        int row = i >> 7, col = i & 127;
        sQ[row * QPAD + col] = qb[i];
    }
    __syncthreads();

    const float* aptr0 = &sQ[(mrow)      * QPAD + 2 * half];   // q rows 0-15
    const float* aptr1 = &sQ[(mrow + 16) * QPAD + 2 * half];   // q rows 16-31
    const float* bbase0 = &sD[0][(tok0 + mrow) * DPAD + 2 * half];
    const float* bbase1 = &sD[1][(tok0 + mrow) * DPAD + 2 * half];

    v8f c0 = {};  // scores[q=0..15][n=0..15]
    v8f c1 = {};  // scores[q=16..31][n=0..15]
#pragma unroll
    for (int kc = 0; kc < D_ / KC; ++kc) {
        const int buf = kc & 1;
        if (kc + 1 < D_ / KC) {
            // WAR guard: in-flight ds reads of the buffer we are about to
            // overwrite (DScnt and ASYNCcnt complete out of order).
            asm volatile("s_wait_dscnt 0" ::: "memory");
            issue_chunk(kc + 1, buf ^ 1);
            asm volatile("s_wait_asynccnt 4" ::: "memory");  // prev 4 landed
        } else {
            asm volatile("s_wait_asynccnt 0" ::: "memory");
        }
        const float* bptr = buf ? bbase1 : bbase0;
        const float* a0p  = aptr0 + kc * KC;
        const float* a1p  = aptr1 + kc * KC;
#pragma unroll
        for (int kk = 0; kk < KC; kk += 4) {
            v2f bf = *(const v2f*)(bptr + kk);   // B frag from LDS (conflict-free)
            v2f a0 = *(const v2f*)(a0p + kk);
            v2f a1 = *(const v2f*)(a1p + kk);
            c0 = __builtin_amdgcn_wmma_f32_16x16x4_f32(false, a0, false, bf,
                                                       (short)0, c0, false, false);
            c1 = __builtin_amdgcn_wmma_f32_16x16x4_f32(false, a1, false, bf,
                                                       (short)0, c1, false, false);
        }
    }

    // Max over the 16 doc tokens (N lives in lanes, per 16-lane half).
    float m0[8], m1[8];
#pragma unroll
    for (int r = 0; r < 8; ++r) { m0[r] = c0[r]; m1[r] = c1[r]; }
#pragma unroll
    for (int off = 8; off >= 1; off >>= 1) {
#pragma unroll
        for (int r = 0; r < 8; ++r) {
            m0[r] = fmaxf(m0[r], __shfl_xor(m0[r], off));
            m1[r] = fmaxf(m1[r], __shfl_xor(m1[r], off));
        }
    }
    // C layout: VGPR r -> M=r (lanes 0-15), M=r+8 (lanes 16-31).
    if (lane == 0) {
#pragma unroll
        for (int r = 0; r < 8; ++r) { sMax[wave][r]      = m0[r]; sMax[wave][16 + r] = m1[r]; }
    } else if (lane == 16) {
#pragma unroll
        for (int r = 0; r < 8; ++r) { sMax[wave][8 + r]  = m0[r]; sMax[wave][24 + r] = m1[r]; }
    }
    __syncthreads();

    // Block-level max over the 8 waves, then one atomic per (b, q).
    if (tid < Q_) {
        float m = sMax[0][tid];
#pragma unroll
        for (int w = 1; w < 8; ++w) m = fmaxf(m, sMax[w][tid]);
        atomicMax(&ws[b * Q_ + tid], encf(m));
    }
}

__global__ void colbert_finalize(const unsigned* __restrict__ ws,
                                 float* __restrict__ out) {
    const int b = blockIdx.x;
    const int t = threadIdx.x;   // 32 threads
    float v = decf(ws[b * Q_ + t]);
#pragma unroll
    for (int off = 16; off >= 1; off >>= 1) v += __shfl_xor(v, off);
    if (t == 0) out[b] = v;
}

extern "C" void kernel_launch(void* const* d_in, const int* in_sizes, int n_in,
                              void* d_out, int out_size, void* d_ws, size_t ws_size,
                              hipStream_t stream) {
    (void)in_sizes; (void)n_in; (void)out_size; (void)ws_size;
    const float* q  = (const float*)d_in[0];
    const float* dm = (const float*)d_in[1];
    float* out      = (float*)d_out;
    unsigned* ws    = (unsigned*)d_ws;

    colbert_init_ws<<<dim3((B_ * Q_ + 255) / 256), 256, 0, stream>>>(ws);
    colbert_maxsim<<<dim3(LD_ / TOK, B_), 256, 0, stream>>>(q, dm, ws);
    colbert_finalize<<<dim3(B_), 32, 0, stream>>>(ws, out);
}